// project_C_spring_boundary_batch_90237262889316
// MI455X (gfx1250) — compile-verified
//
#include <hip/hip_runtime.h>
#include <hip/hip_bf16.h>
#include <math.h>

// Problem constants (from reference)
constexpr int B  = 8;
constexpr int NV = 8192;
constexpr int E  = 32768;
constexpr int MROWS  = 32;   // 24 real rows (b*3+c) padded to 32 (two 16-row WMMA tiles)
constexpr int KSPLIT = 8;    // K-dimension split for occupancy
constexpr int KSLICE = E / KSPLIT;      // 4096
constexpr int NTILES = NV / 16;         // 512

typedef __attribute__((ext_vector_type(2))) float v2f;
typedef __attribute__((ext_vector_type(4))) float v4f;
typedef __attribute__((ext_vector_type(8))) float v8f;

// Packed A layout (WMMA-native, one b128 per lane per K-step):
//   Apk[(e>>1)*64 + col*4 + (e&1) + 2*(row>=16)] = M[e][row], col = row&15
// i.e. per (K-pair, col) a float4 { M[k][col], M[k+1][col], M[k][col+16], M[k+1][col+16] }

// ---------------------------------------------------------------------------
// Kernel 1: per-edge XPBD math. Writes L_new (output tail) and packed A.
// ---------------------------------------------------------------------------
__global__ void edge_kernel(const float* __restrict__ Vp,   // (B,NV,3)
                            const float* __restrict__ L,    // (B,E)
                            const float* __restrict__ Vw,   // (B,NV)
                            const float* __restrict__ Vc,   // (B)
                            const int*   __restrict__ Cd,   // (E,2)
                            const float* __restrict__ Ci,   // (E)
                            float* __restrict__ Apk,        // (E/2,16,4) floats
                            float* __restrict__ Lnew)       // (B,E)
{
    int t = blockIdx.x * blockDim.x + threadIdx.x;   // t in [0, B*E)
    int e = t & (E - 1);
    int b = t >> 15;                                 // t / E

    int i = Cd[2 * e];
    int j = Cd[2 * e + 1];

    const float* pi = Vp + ((size_t)b * NV + i) * 3;
    const float* pj = Vp + ((size_t)b * NV + j) * 3;
    float nx = pi[0] - pj[0];
    float ny = pi[1] - pj[1];
    float nz = pi[2] - pj[2];
    float D  = sqrtf(nx * nx + ny * ny + nz * nz);
    float C  = D - Ci[e];
    float inv = 1.0f / (D + 1e-8f);

    float A = Vc[b];
    float S = Vw[(size_t)b * NV + i] + Vw[(size_t)b * NV + j];
    if (S == 0.0f) S = INFINITY;

    float l  = L[(size_t)b * E + e];
    float ld = (-C - A * l) / (S + A);

    Lnew[(size_t)b * E + e] = l + ld;

    float s = ld * inv;
    float m[3] = { s * nx, s * ny, s * nz };

    const size_t base = (size_t)(e >> 1) * 64 + (e & 1);
    #pragma unroll
    for (int c = 0; c < 3; ++c) {
        int r = b * 3 + c;
        Apk[base + (size_t)(r & 15) * 4 + ((r >> 4) << 1)] = m[c];
    }

    if (b == 0) {            // zero padding rows 24..31 (cols 8..15, hi slot)
        #pragma unroll
        for (int col = 8; col < 16; ++col)
            Apk[base + (size_t)col * 4 + 2] = 0.0f;
    }
}

// ---------------------------------------------------------------------------
// Kernel 2: P[ks][m][n] = sum_{k in slice ks} M[k][m] * C_mtx[k][n]
// One wave per (16-wide n-tile, K-slice); two 16x16 f32 accumulators
// (m=0..15, 16..31) share every B load. V_WMMA_F32_16X16X4_F32.
//
// A (16x4 f32): lane<16 -> M=lane, V0=K0,V1=K1; lane>=16 -> M=lane-16, V0=K2,V1=K3
// D (16x16 f32): VGPR r, lane<16 -> M=r, lane>=16 -> M=r+8, N=lane%16
// ---------------------------------------------------------------------------
__global__ void __launch_bounds__(256)
gemm_partial(const float* __restrict__ Cm,      // (E,NV)
             const v4f*   __restrict__ Apk,     // (E/2,16) float4
             float* __restrict__ P)             // (KSPLIT,32,NV)
{
    const int lane = threadIdx.x & 31;
    const int wid  = threadIdx.x >> 5;
    const int tile = blockIdx.x * 8 + wid;       // 8 waves per block
    const int ntile = tile & (NTILES - 1);
    const int ks    = tile / NTILES;

    const int col  = lane & 15;                  // N within tile / M within tile
    const int h    = lane >> 4;                  // lane half
    const int koff = h << 1;                     // 0 or 2
    const int n0   = ntile * 16;
    const int k0   = ks * KSLICE;
    const int k1   = k0 + KSLICE;

    v8f c0 = {};   // m rows 0..15
    v8f c1 = {};   // m rows 16..31

    const float* bptr = Cm + (size_t)(k0 + koff) * NV + n0 + col;
    const v4f*   aptr = Apk + (size_t)(k0 / 2 + h) * 16 + col;

    #pragma unroll 4
    for (int k = k0; k < k1; k += 4) {
        v2f bv, a0, a1;
        bv[0] = bptr[0];
        bv[1] = bptr[NV];
        v4f av = *aptr;
        a0[0] = av.x; a0[1] = av.y;
        a1[0] = av.z; a1[1] = av.w;
        bptr += (size_t)4 * NV;
        aptr += 2 * 16;
        c0 = __builtin_amdgcn_wmma_f32_16x16x4_f32(false, a0, false, bv,
                                                   (short)0, c0, false, false);
        c1 = __builtin_amdgcn_wmma_f32_16x16x4_f32(false, a1, false, bv,
                                                   (short)0, c1, false, false);
    }

    const int mbase = h * 8;
    float* Pks = P + (size_t)ks * MROWS * NV;
    #pragma unroll
    for (int r = 0; r < 8; ++r) {
        Pks[(size_t)(mbase + r) * NV + n0 + col]      = c0[r];
        Pks[(size_t)(mbase + r + 16) * NV + n0 + col] = c1[r];
    }
}

// ---------------------------------------------------------------------------
// Kernel 3: reduce K-partials, apply V_predict_new = V_predict + V_w * L_vert
// ---------------------------------------------------------------------------
__global__ void finalize(const float* __restrict__ Vp,   // (B,NV,3)
                         const float* __restrict__ Vw,   // (B,NV)
                         const float* __restrict__ P,    // (KSPLIT,32,NV)
                         float* __restrict__ out)        // (B,NV,3)
{
    int t = blockIdx.x * blockDim.x + threadIdx.x;       // [0, B*NV*3)
    if (t >= B * NV * 3) return;
    int c = t % 3;
    int n = (t / 3) & (NV - 1);
    int b = t / (3 * NV);
    int m = b * 3 + c;

    float s = 0.0f;
    #pragma unroll
    for (int ks = 0; ks < KSPLIT; ++ks)
        s += P[((size_t)ks * MROWS + m) * NV + n];

    out[t] = Vp[t] + Vw[(size_t)b * NV + n] * s;
}

// ---------------------------------------------------------------------------
extern "C" void kernel_launch(void* const* d_in, const int* in_sizes, int n_in,
                              void* d_out, int out_size, void* d_ws, size_t ws_size,
                              hipStream_t stream) {
    const float* Vp = (const float*)d_in[0];   // V_predict (B,NV,3)
    const float* L  = (const float*)d_in[1];   // L (B,E,1)
    const float* Vw = (const float*)d_in[2];   // V_w (B,NV,1)
    const float* Vc = (const float*)d_in[3];   // V_compliance (B,1,1)
    const int*   Cd = (const int*)d_in[4];     // C_dist (E,2)
    const float* Ci = (const float*)d_in[5];   // C_init_d (E,1)
    const float* Cm = (const float*)d_in[6];   // C_mtx (E,NV)

    float* out  = (float*)d_out;
    float* Lnew = out + (size_t)B * NV * 3;               // L_new tail of d_out
    float* Apk  = (float*)d_ws;                           // E*32 floats = 4 MB
    float* P    = Apk + (size_t)E * MROWS;                // KSPLIT*32*NV = 8 MB

    edge_kernel<<<(B * E) / 256, 256, 0, stream>>>(Vp, L, Vw, Vc, Cd, Ci, Apk, Lnew);
    gemm_partial<<<(NTILES * KSPLIT) / 8, 256, 0, stream>>>(Cm, (const v4f*)Apk, P);
    finalize<<<(B * NV * 3 + 255) / 256, 256, 0, stream>>>(Vp, Vw, P, out);
}